// VideoDiffusionLossPnP_80401787781460
// MI455X (gfx1250) — compile-verified
//
#include <hip/hip_runtime.h>
#include <math.h>

typedef __attribute__((ext_vector_type(2))) float v2f;
typedef __attribute__((ext_vector_type(4))) float v4f;
typedef __attribute__((ext_vector_type(8))) float v8f;

// ---- static problem constants (mirror the reference) ----
#define BZ        8
#define LMAX      5500
#define DTOK      64
#define TXT       225
#define NFT       900          // tokens per latent frame (30*30)
#define IMG_LEN   1800         // 2 frames * 900
#define TOK_ITEM  3600         // tokens consumed per batch item
#define SEG0_IMG  225          // seg0: txt [0,225), img [225,2025)
#define SEG0_END  2025
#define SEG1_IMG  2250         // seg1: txt [2025,2250), img [2250,4050)
#define SEG1_END  4050

// flat f32 offsets into d_out (outputs concatenated in return order)
#define OFF_LOSS  0
#define OFF_TOK   8
#define OFF_TXT   2816008
#define OFF_IMG   2860008
#define OFF_POS   2904008
#define OFF_ATTN  3036008

#define BLOCKS_PER_B 225       // 3600 tokens * 16 chan / 256 threads
#define INV_ELEMS   (1.0f / 352000.0f)   // mean over L*D = 5500*64

// 25 divides 2025, 4050, 5500 -> one mask pattern per 25-row chunk
#define ROWS_PER_BLOCK 25
#define CHUNKS_PER_B   (LMAX / ROWS_PER_BLOCK)   // 220

#if __has_builtin(__builtin_amdgcn_global_store_async_from_lds_b128) && \
    __has_builtin(__builtin_amdgcn_s_wait_asynccnt)
#define HAVE_ASYNC_LDS 1
// builtin signature (from hipcc diagnostic): pointers to int32x4 in
// addrspace(1) (global) and addrspace(3) (LDS), plus imm offset + imm cpol
typedef int i32x4 __attribute__((ext_vector_type(4)));
typedef __attribute__((address_space(1))) i32x4 glb_i32x4;
typedef __attribute__((address_space(3))) i32x4 lds_i32x4;
#endif

// ---------------------------------------------------------------------------
// txt_masks / image_masks / position_ids  (tiny: ~0.9 MB)
// ---------------------------------------------------------------------------
__global__ void mask_pos_kernel(float* __restrict__ out) {
    int R = blockIdx.x * blockDim.x + threadIdx.x;
    if (R >= BZ * LMAX) return;
    int l = R % LMAX;
    bool txt = (l < TXT) || (l >= SEG0_END && l < SEG1_IMG);
    bool img = (l >= SEG0_IMG && l < SEG0_END) || (l >= SEG1_IMG && l < SEG1_END);
    float fi = -1.f, fj = -1.f, fk = -1.f;
    if (img) {
        int m = (l < SEG0_END) ? (l - SEG0_IMG) : (l - SEG1_IMG);
        int i = m / NFT;
        int r = m - i * NFT;
        int j = r / 30;
        int k = r - j * 30;
        fi = (float)i; fj = (float)j; fk = (float)k;
    }
    out[OFF_TXT + R] = txt ? 1.f : 0.f;
    out[OFF_IMG + R] = img ? 1.f : 0.f;
    float* pos = out + OFF_POS + (size_t)R * 3;
    pos[0] = fi; pos[1] = fj; pos[2] = fk;
}

// ---------------------------------------------------------------------------
// zero the non-image token regions: l in [0,225) U [2025,2250) U [4050,5500)
// per item that is 1900*64 = 121600 floats = 30400 float4
// ---------------------------------------------------------------------------
__global__ void zero_tok_kernel(float* __restrict__ out) {
    int idx = blockIdx.x * blockDim.x + threadIdx.x;
    if (idx >= BZ * 30400) return;
    int b = idx / 30400;
    int f = (idx - b * 30400) * 4;          // float offset inside zero-space
    int off;
    if (f < 14400)            off = f;                      // [0,225)*64
    else if (f < 28800)       off = 129600 + (f - 14400);   // [2025,2250)*64
    else                      off = 259200 + (f - 28800);   // [4050,5500)*64
    v4f z = {0.f, 0.f, 0.f, 0.f};
    float* tok = out + OFF_TOK + (size_t)b * (LMAX * DTOK) + off;
    __builtin_nontemporal_store(z, (v4f*)tok);
}

// ---------------------------------------------------------------------------
// fused patchify -> tokens + weighted-MSE partial sums (deterministic)
// thread = (b, src_token, channel c); handles the 4 (p,q) patch elements
// token[(b,l), (2p+q)*16+c] = input[0, t, c, 2hh+p, 2ww+q]
// ---------------------------------------------------------------------------
__global__ void tok_loss_kernel(const float* __restrict__ in,
                                const float* __restrict__ noise,
                                const float* __restrict__ alphas,
                                float* __restrict__ out,
                                float* __restrict__ partial) {
    int bid = blockIdx.x;                               // 0..1799
    int b   = bid / BLOCKS_PER_B;
    int tin = (bid % BLOCKS_PER_B) * 256 + threadIdx.x; // 0..57599
    int src = tin >> 4;                                 // token within item
    int c   = tin & 15;                                 // latent channel
    int l   = (src < IMG_LEN) ? (SEG0_IMG + src) : (SEG1_IMG + (src - IMG_LEN));
    int g   = b * TOK_ITEM + src;                       // global x-token
    int t16 = g / NFT;                                  // latent frame t (0..31)
    int r   = g - t16 * NFT;
    int hh  = r / 30;
    int ww  = r - hh * 30;

    const float* ip = in + ((size_t)(t16 * 16 + c) * 60 + hh * 2) * 60 + ww * 2;
    float v00 = ip[0], v01 = ip[1], v10 = ip[60], v11 = ip[61];

    size_t tokBase = ((size_t)(b * LMAX + l)) * DTOK + c;
    float* tok = out + OFF_TOK;
    tok[tokBase +  0] = v00;   // p=0,q=0 -> d = c
    tok[tokBase + 16] = v01;   // p=0,q=1 -> d = 16+c
    tok[tokBase + 32] = v10;   // p=1,q=0 -> d = 32+c
    tok[tokBase + 48] = v11;   // p=1,q=1 -> d = 48+c

    float a     = alphas[b];
    float om    = 1.0f - a * a;
    float sig   = sqrtf(om);
    float scale = INV_ELEMS / om;                       // wgt folded with mean
    float am1   = a - 1.0f;

    const float* nz = noise + tokBase;
    float d0 = v00 * am1 + nz[0]  * sig;
    float d1 = v01 * am1 + nz[16] * sig;
    float d2 = v10 * am1 + nz[32] * sig;
    float d3 = v11 * am1 + nz[48] * sig;
    float acc = scale * (d0*d0 + d1*d1 + d2*d2 + d3*d3);

    // wave32 reduce, then cross-wave via LDS (fixed order -> deterministic)
    #pragma unroll
    for (int off = 16; off > 0; off >>= 1) acc += __shfl_down(acc, off, 32);
    __shared__ float smem[8];
    int lane = threadIdx.x & 31, wid = threadIdx.x >> 5;
    if (lane == 0) smem[wid] = acc;
    __syncthreads();
    if (threadIdx.x == 0) {
        float s = 0.f;
        #pragma unroll
        for (int i = 0; i < 8; ++i) s += smem[i];
        partial[bid] = s;
    }
}

// ---------------------------------------------------------------------------
// final loss reduce: 1 wave per batch item; the 32 lane-partials are reduced
// with V_WMMA_F32_16X16X4_F32 against a ones B-matrix:
//   A[m,0]=s_m (lanes 0-15, VGPR0), A[m,2]=s_{m+16} (lanes 16-31, VGPR0),
//   A[m,1]=A[m,3]=0  ->  D[m,n] = s_m + s_{m+16} for all n.
// ---------------------------------------------------------------------------
__global__ void loss_finish_kernel(const float* __restrict__ partial,
                                   float* __restrict__ out) {
    int b    = blockIdx.x;
    int lane = threadIdx.x;                 // 0..31 (wave32)
    float s = 0.f;
    #pragma unroll
    for (int k = 0; k < 8; ++k) {
        int idx = lane + 32 * k;            // 225 partials per item
        int cl  = (idx < BLOCKS_PER_B) ? idx : (BLOCKS_PER_B - 1);
        float v = partial[b * BLOCKS_PER_B + cl];
        s += (idx < BLOCKS_PER_B) ? v : 0.0f;
    }
#if __has_builtin(__builtin_amdgcn_wmma_f32_16x16x4_f32)
    v2f A;  A.x = s;   A.y = 0.0f;
    v2f Bv; Bv.x = 1.0f; Bv.y = 1.0f;       // 4x16 all-ones
    v8f Cv = {};
    v8f Dv = __builtin_amdgcn_wmma_f32_16x16x4_f32(
        /*neg_a=*/false, A, /*neg_b=*/false, Bv,
        /*c_mod=*/(short)0, Cv, /*reuse_a=*/false, /*reuse_b=*/false);
    float t = Dv[0] + Dv[1] + Dv[2] + Dv[3] + Dv[4] + Dv[5] + Dv[6] + Dv[7];
    float total = t + __shfl_xor(t, 16, 32);
    if (lane == 0) out[OFF_LOSS + b] = total;
#else
    #pragma unroll
    for (int off = 16; off > 0; off >>= 1) s += __shfl_down(s, off, 32);
    if (lane == 0) out[OFF_LOSS + b] = s;
#endif
}

// ---------------------------------------------------------------------------
// attention mask: 968 MB of streaming writes -> the whole kernel's roofline.
// The output is 3 distinct 22KB row patterns replicated. Per block: build the
// pattern once in LDS, then stream 25 rows out with the CDNA5 async
// LDS->memory path (GLOBAL_STORE_ASYNC_FROM_LDS_B128, ASYNCcnt-tracked) so
// the wave only generates addresses. Chunk boundaries (2025, 4050) are
// multiples of 25, so a chunk never straddles two patterns.
// Fallback: LDS read + b128 non-temporal direct stores.
// ---------------------------------------------------------------------------
__global__ void attn_kernel(float* __restrict__ out) {
    __shared__ __align__(16) float rowbuf[LMAX];     // 22000 B
    int chunk = blockIdx.x;                          // 0 .. BZ*220-1
    int b     = chunk / CHUNKS_PER_B;
    int row0  = (chunk - b * CHUNKS_PER_B) * ROWS_PER_BLOCK;

    // pattern for this chunk (uniform across the block)
    int lo, hi;
    if (row0 < SEG0_END)      { lo = 0;        hi = SEG0_END; }
    else if (row0 < SEG1_END) { lo = SEG0_END; hi = SEG1_END; }
    else                      { lo = 0;        hi = 0; }

    // build the 5500-float row pattern in LDS once
    for (int j = threadIdx.x; j < LMAX; j += blockDim.x)
        rowbuf[j] = (j >= lo && j < hi) ? 1.f : 0.f;
    __syncthreads();

    float* base = out + (size_t)OFF_ATTN + ((size_t)b * LMAX + row0) * LMAX;
    for (int rr = 0; rr < ROWS_PER_BLOCK; ++rr) {
        float* rp = base + (size_t)rr * LMAX;
#if defined(HAVE_ASYNC_LDS)
        for (int j = threadIdx.x; j < LMAX / 4; j += blockDim.x) {
            __builtin_amdgcn_global_store_async_from_lds_b128(
                (glb_i32x4*)(rp + j * 4), (lds_i32x4*)(rowbuf + j * 4),
                /*offset=*/0, /*cpol=*/0);
        }
#else
        for (int j = threadIdx.x; j < LMAX / 4; j += blockDim.x) {
            v4f v = *(const v4f*)(rowbuf + j * 4);
            __builtin_nontemporal_store(v, (v4f*)(rp + j * 4));
        }
#endif
    }
#if defined(HAVE_ASYNC_LDS)
    __builtin_amdgcn_s_wait_asynccnt(0);   // drain before wave exit
#endif
}

// ---------------------------------------------------------------------------
extern "C" void kernel_launch(void* const* d_in, const int* in_sizes, int n_in,
                              void* d_out, int out_size, void* d_ws, size_t ws_size,
                              hipStream_t stream) {
    const float* input  = (const float*)d_in[0];   // [1,32,16,60,60]
    const float* noise  = (const float*)d_in[1];   // [8,5500,64]
    const float* alphas = (const float*)d_in[2];   // [8]
    float* out     = (float*)d_out;
    float* partial = (float*)d_ws;                 // 1800 floats of scratch

    mask_pos_kernel   <<<(BZ * LMAX + 255) / 256, 256, 0, stream>>>(out);
    zero_tok_kernel   <<<(BZ * 30400 + 255) / 256, 256, 0, stream>>>(out);
    tok_loss_kernel   <<<BZ * BLOCKS_PER_B, 256, 0, stream>>>(input, noise, alphas, out, partial);
    loss_finish_kernel<<<BZ, 32, 0, stream>>>(partial, out);
    attn_kernel       <<<BZ * CHUNKS_PER_B, 256, 0, stream>>>(out);
}